// fMRIPrior_18150531793332
// MI455X (gfx1250) — compile-verified
//
#include <hip/hip_runtime.h>
#include <hip/hip_bf16.h>

typedef __attribute__((ext_vector_type(2))) float v2f;
typedef __attribute__((ext_vector_type(4))) float v4f;
typedef __attribute__((ext_vector_type(8))) float v8f;

#define B_  32
#define C_  64
#define H_  224
#define W_  224
#define L_  68
#define RAD_ 45.0f
#define INV2S2 (1.0f / (2.0f * 15.0f * 15.0f))

#define TILES_H (H_ / 16)            // 14
#define TILES_W (W_ / 16)            // 14
#define TILES_PER_B (TILES_H * TILES_W)   // 196
#define N_TILES (B_ * TILES_PER_B)        // 6272
#define WAVES_PER_BLOCK 4

// ---------------- Kernel 1: separable Gaussian factors ----------------
// gy[b,l,h] = mask(dy) * exp(-dy^2/(2s^2)), dy = h - trunc(lm_y)
// gx[b,l,w] analogous with lm_x.
__global__ void gauss_kernel(const float* __restrict__ lm,
                             float* __restrict__ gy,
                             float* __restrict__ gx) {
    const int NG = B_ * L_ * H_;          // == B_*L_*W_
    int i = blockIdx.x * blockDim.x + threadIdx.x;
    if (i < NG) {
        int h  = i % H_;
        int bl = i / H_;
        float ly = truncf(lm[bl * 2 + 1]);
        float d  = (float)h - ly;
        float v  = (d >= -RAD_ && d < RAD_) ? __expf(-(d * d) * INV2S2) : 0.0f;
        gy[i] = v;
    } else if (i < 2 * NG) {
        int j  = i - NG;
        int w  = j % W_;
        int bl = j / W_;
        float lx = truncf(lm[bl * 2 + 0]);
        float d  = (float)w - lx;
        float v  = (d >= -RAD_ && d < RAD_) ? __expf(-(d * d) * INV2S2) : 0.0f;
        gx[j] = v;
    }
}

// ---------------- Kernel 2: batched GEMM via V_WMMA_F32_16X16X4_F32 ----
// prior[b, h, w] = sum_l gy[b,l,h] * gx[b,l,w]
// One wave per 16x16 tile. A = gy^T tile (16x4 per step), B = gx tile (4x16).
// Fragment layout (ISA 7.12.2, 32-bit A 16x4 / B 4x16):
//   vgpr j holds K = j + 2*(lane>>4); row/col index = lane & 15.
__global__ void __launch_bounds__(WAVES_PER_BLOCK * 32)
gemm_wmma_kernel(const float* __restrict__ gy,
                 const float* __restrict__ gx,
                 float* __restrict__ prior) {
    const int lane = threadIdx.x & 31;
    const int wib  = threadIdx.x >> 5;
    const int wid  = blockIdx.x * WAVES_PER_BLOCK + wib;   // exact: no guard -> EXEC all 1s

    const int b  = wid / TILES_PER_B;
    const int t  = wid % TILES_PER_B;
    const int th = t / TILES_W;
    const int tw = t % TILES_W;

    const int m   = lane & 15;      // row (A) / col (B) index within tile
    const int khi = lane >> 4;      // selects K pair {0,1} vs {2,3}

    const float* gyb = gy + (b * L_) * H_ + th * 16;  // + l*H_ + m
    const float* gxb = gx + (b * L_) * W_ + tw * 16;  // + l*W_ + n

    v8f c = {};
#pragma unroll
    for (int k0 = 0; k0 < L_; k0 += 4) {
        v2f a, bb;
        const int kA = k0 + 2 * khi;
        a.x  = gyb[(kA + 0) * H_ + m];
        a.y  = gyb[(kA + 1) * H_ + m];
        bb.x = gxb[(kA + 0) * W_ + m];
        bb.y = gxb[(kA + 1) * W_ + m];
        c = __builtin_amdgcn_wmma_f32_16x16x4_f32(
                /*neg_a=*/false, a, /*neg_b=*/false, bb,
                /*c_mod=*/(short)0, c, /*reuse_a=*/false, /*reuse_b=*/false);
    }

    // C/D layout: VGPR r, lanes 0-15 -> M=r, lanes 16-31 -> M=r+8; N = lane&15
    float* pb = prior + (b * H_ + th * 16) * W_ + tw * 16;
    const int mrow = khi * 8;
    const int n    = m;
#pragma unroll
    for (int r = 0; r < 8; ++r) {
        pb[(r + mrow) * W_ + n] = c[r];
    }
}

// ---------------- Kernel 3/4: deterministic global max -> scale --------
__global__ void reduce_max1(const float* __restrict__ prior,
                            float* __restrict__ partial, int n4) {
    __shared__ float sm[256];
    const v4f* p4 = (const v4f*)prior;
    float m = 0.0f;   // prior >= 0
    for (int i = blockIdx.x * blockDim.x + threadIdx.x; i < n4;
         i += gridDim.x * blockDim.x) {
        v4f v = p4[i];
        m = fmaxf(m, fmaxf(fmaxf(v.x, v.y), fmaxf(v.z, v.w)));
    }
    sm[threadIdx.x] = m;
    __syncthreads();
    for (int s = 128; s > 0; s >>= 1) {
        if (threadIdx.x < s) sm[threadIdx.x] = fmaxf(sm[threadIdx.x], sm[threadIdx.x + s]);
        __syncthreads();
    }
    if (threadIdx.x == 0) partial[blockIdx.x] = sm[0];
}

__global__ void reduce_max2(const float* __restrict__ partial,
                            float* __restrict__ scale, int n) {
    __shared__ float sm[256];
    float m = 0.0f;
    for (int i = threadIdx.x; i < n; i += 256) m = fmaxf(m, partial[i]);
    sm[threadIdx.x] = m;
    __syncthreads();
    for (int s = 128; s > 0; s >>= 1) {
        if (threadIdx.x < s) sm[threadIdx.x] = fmaxf(sm[threadIdx.x], sm[threadIdx.x + s]);
        __syncthreads();
    }
    if (threadIdx.x == 0) *scale = 1.0f / (sm[0] + 1e-8f);
}

// ---------------- Kernel 5: bandwidth-bound apply ----------------------
// out[b,c,h,w] = x[b,c,h,w] * prior[b,h,w] * scale   (float4 streams)
__global__ void __launch_bounds__(256)
apply_kernel(const float* __restrict__ x,
             const float* __restrict__ prior,
             const float* __restrict__ scale,
             float* __restrict__ out, int n4) {
    int i = blockIdx.x * blockDim.x + threadIdx.x;
    if (i >= n4) return;
    const float s = *scale;
    const v4f xv = ((const v4f*)x)[i];
    const int e  = i << 2;
    const int b  = e / (C_ * H_ * W_);
    const int hw = e % (H_ * W_);
    const v4f pv = ((const v4f*)prior)[(b * (H_ * W_) + hw) >> 2];
    v4f r;
    r.x = xv.x * (pv.x * s);
    r.y = xv.y * (pv.y * s);
    r.z = xv.z * (pv.z * s);
    r.w = xv.w * (pv.w * s);
    ((v4f*)out)[i] = r;
}

extern "C" void kernel_launch(void* const* d_in, const int* in_sizes, int n_in,
                              void* d_out, int out_size, void* d_ws, size_t ws_size,
                              hipStream_t stream) {
    const float* x  = (const float*)d_in[0];   // (B,C,H,W) fp32
    const float* lm = (const float*)d_in[1];   // (B,L,2)   fp32
    float* out = (float*)d_out;

    // Workspace layout (floats), 16B-aligned slices
    float* ws = (float*)d_ws;
    const size_t NGY = (size_t)B_ * L_ * H_;       // 487,424
    const size_t NGX = (size_t)B_ * L_ * W_;       // 487,424
    const size_t NP  = (size_t)B_ * H_ * W_;       // 1,605,632
    float* gy      = ws;
    float* gx      = gy + NGY;
    float* prior   = gx + NGX;
    float* partial = prior + NP;                   // 1024 entries
    float* scale   = partial + 1024;

    // 1) Gaussian factors
    {
        int total = (int)(NGY + NGX);
        int blocks = (total + 255) / 256;
        gauss_kernel<<<blocks, 256, 0, stream>>>(lm, gy, gx);
    }
    // 2) Batched GEMM via WMMA (one wave per 16x16 tile)
    {
        int blocks = N_TILES / WAVES_PER_BLOCK;    // 1568, exact
        gemm_wmma_kernel<<<blocks, WAVES_PER_BLOCK * 32, 0, stream>>>(gy, gx, prior);
    }
    // 3) Global max, pass 1 (grid-stride over float4)
    {
        int n4 = (int)(NP / 4);
        reduce_max1<<<1024, 256, 0, stream>>>(prior, partial, n4);
    }
    // 4) Global max, pass 2 -> scale = 1/(max + 1e-8)
    reduce_max2<<<1, 256, 0, stream>>>(partial, scale, 1024);
    // 5) Apply (bandwidth bound: ~822 MB -> ~35 us at 23.3 TB/s)
    {
        int n4 = (B_ * C_ * H_ * W_) / 4;          // 25,690,112
        int blocks = (n4 + 255) / 256;
        apply_kernel<<<blocks, 256, 0, stream>>>(x, prior, scale, out, n4);
    }
}